// Attention_65403761984268
// MI455X (gfx1250) — compile-verified
//
#include <hip/hip_runtime.h>
#include <math.h>

// ---------------------------------------------------------------------------
// Attention forward for MI455X (gfx1250), fp32 via V_WMMA_F32_16X16X4_F32.
// B=4, N=4096, DIM_IN=DIM_INNER=DIM_OUT=512.
// Workgroup tile: 64 rows x 128 cols, 8 waves, each wave 16x64 (4 accums).
// LDS: A row-major [64][20] (padded), B transposed [128][20] (padded) so every
// WMMA fragment is one contiguous ds_load_b64.
// ---------------------------------------------------------------------------

typedef __attribute__((ext_vector_type(2))) float v2f;
typedef __attribute__((ext_vector_type(8))) float v8f;

#define WMMA_F32(a, b, c) \
  __builtin_amdgcn_wmma_f32_16x16x4_f32(false, (a), false, (b), (short)0, (c), false, false)

static constexpr int   BATCH   = 4;
static constexpr int   SEQ     = 4096;
static constexpr int   DIM     = 512;
static constexpr int   THREE_D = 1536;
static constexpr int   LP      = 20;                      // padded LDS stride
static constexpr float SCALE   = 0.04419417382415922f;    // 512^-0.5
static constexpr float NEGMAX  = -3.4028234663852886e38f; // -FLT_MAX

// Async global->LDS copy (GLOBAL_LOAD_ASYNC_TO_LDS_B128, tracked by ASYNCcnt).
__device__ __forceinline__ void async_copy_b128(void* lds_dst, const float* gsrc) {
  asm volatile("global_load_async_to_lds_b128 %0, %1, off"
               :: "v"((unsigned)(unsigned long long)lds_dst), "v"(gsrc)
               : "memory");
}
__device__ __forceinline__ void wait_async_zero() {
  asm volatile("s_wait_asynccnt 0x0" ::: "memory");
}

// ---------------------------------------------------------------------------
// Kernel 1: QKV projection. C[16384x1536] = X[16384x512] @ Wqkv[512x1536],
// de-interleaved into contiguous Q, K, V ([16384 x 512] each).
// A tile is filled via async global->LDS copies.
// ---------------------------------------------------------------------------
__global__ __launch_bounds__(256) void qkv_gemm(
    const float* __restrict__ X, const float* __restrict__ W,
    float* __restrict__ Qb, float* __restrict__ Kb, float* __restrict__ Vb) {
  __shared__ __align__(16) float sA[64][LP];
  __shared__ __align__(16) float sBT[128][LP];
  const int row0 = blockIdx.y * 64;
  const int col0 = blockIdx.x * 128;
  const int t    = threadIdx.x;
  const int lane = t & 31;
  const int wave = t >> 5;        // 0..7
  const int wr   = wave >> 1;     // 0..3
  const int wc   = wave & 1;      // 0..1
  const int lrow = lane & 15;
  const int lhi  = lane >> 4;     // 0/1

  const int ar = t >> 2;          // A tile: 64 rows x 16 k
  const int ac = (t & 3) * 4;
  const int kB = t >> 4;          // B tile: 16 k x 128 n
  const int nB = (t & 15) * 8;

  v8f acc[4] = {};
  for (int k0 = 0; k0 < DIM; k0 += 16) {
    if (k0 + 16 < DIM) {
      __builtin_prefetch(X + (size_t)(row0 + ar) * DIM + k0 + 16 + ac, 0, 1);
      __builtin_prefetch(W + (size_t)(k0 + 16 + kB) * THREE_D + col0 + nB, 0, 1);
    }
    // A: async 16B copies straight into LDS (one per thread).
    async_copy_b128(&sA[ar][ac], X + (size_t)(row0 + ar) * DIM + k0 + ac);
    // B: load row-major [k][n], scatter transposed into sBT[n][k].
    float4 w0 = *(const float4*)(W + (size_t)(k0 + kB) * THREE_D + col0 + nB);
    float4 w1 = *(const float4*)(W + (size_t)(k0 + kB) * THREE_D + col0 + nB + 4);
    sBT[nB + 0][kB] = w0.x; sBT[nB + 1][kB] = w0.y;
    sBT[nB + 2][kB] = w0.z; sBT[nB + 3][kB] = w0.w;
    sBT[nB + 4][kB] = w1.x; sBT[nB + 5][kB] = w1.y;
    sBT[nB + 6][kB] = w1.z; sBT[nB + 7][kB] = w1.w;
    wait_async_zero();
    __syncthreads();
#pragma unroll
    for (int kk = 0; kk < 16; kk += 4) {
      const int ka = kk + 2 * lhi;
      v2f a;
      a.x = sA[16 * wr + lrow][ka];
      a.y = sA[16 * wr + lrow][ka + 1];
#pragma unroll
      for (int cc = 0; cc < 4; ++cc) {
        v2f bf;
        bf.x = sBT[64 * wc + 16 * cc + lrow][ka];
        bf.y = sBT[64 * wc + 16 * cc + lrow][ka + 1];
        acc[cc] = WMMA_F32(a, bf, acc[cc]);
      }
    }
    __syncthreads();
  }
  const int grow_base = row0 + 16 * wr + 8 * lhi;
  const int gcol      = col0 + 64 * wc + lrow;
#pragma unroll
  for (int r = 0; r < 8; ++r) {
    const size_t off = (size_t)(grow_base + r) * DIM;
#pragma unroll
    for (int cc = 0; cc < 4; ++cc) {
      const int c = gcol + 16 * cc;
      const float v = acc[cc][r];
      if (c < DIM)          Qb[off + c]           = v;
      else if (c < 2 * DIM) Kb[off + c - DIM]     = v;
      else                  Vb[off + c - 2 * DIM] = v;
    }
  }
}

// ---------------------------------------------------------------------------
// Kernel 2: sim = Q @ K^T * scale + prev, causal mask -> -FLT_MAX.
// Per batch: M=N=4096, K=512. Fully masked tiles skip the GEMM entirely.
// K tile ([j][k] row-major) is exactly the transposed-LDS layout: no scatter.
// ---------------------------------------------------------------------------
__global__ __launch_bounds__(256) void sim_gemm(
    const float* __restrict__ Qb, const float* __restrict__ Kb,
    const float* __restrict__ prev, float* __restrict__ sim) {
  const int b    = blockIdx.z;
  const int row0 = blockIdx.y * 64;    // i
  const int col0 = blockIdx.x * 128;   // j
  const float* Q = Qb   + (size_t)b * SEQ * DIM;
  const float* K = Kb   + (size_t)b * SEQ * DIM;
  const float* P = prev + (size_t)b * SEQ * SEQ;
  float*       S = sim  + (size_t)b * SEQ * SEQ;
  const int t = threadIdx.x;

  if (col0 >= row0 + 64) {  // tile strictly above the diagonal: pure mask fill
    for (int e = t; e < 64 * 128; e += 256) {
      const int r = e >> 7, c = e & 127;
      S[(size_t)(row0 + r) * SEQ + col0 + c] = NEGMAX;
    }
    return;
  }

  __shared__ __align__(16) float sA[64][LP];
  __shared__ __align__(16) float sBT[128][LP];
  const int lane = t & 31;
  const int wave = t >> 5;
  const int wr   = wave >> 1;
  const int wc   = wave & 1;
  const int lrow = lane & 15;
  const int lhi  = lane >> 4;

  const int ar = t >> 2;          // A tile 64x16 (Q rows)
  const int ac = (t & 3) * 4;
  const int jr = t >> 1;          // B tile: 128 K-rows (j) x 16 dims
  const int jc = (t & 1) * 8;

  v8f acc[4] = {};
  for (int k0 = 0; k0 < DIM; k0 += 16) {
    float4 va = *(const float4*)(Q + (size_t)(row0 + ar) * DIM + k0 + ac);
    *(float4*)&sA[ar][ac] = va;
    float4 vk0 = *(const float4*)(K + (size_t)(col0 + jr) * DIM + k0 + jc);
    float4 vk1 = *(const float4*)(K + (size_t)(col0 + jr) * DIM + k0 + jc + 4);
    *(float4*)&sBT[jr][jc]     = vk0;
    *(float4*)&sBT[jr][jc + 4] = vk1;
    __syncthreads();
#pragma unroll
    for (int kk = 0; kk < 16; kk += 4) {
      const int ka = kk + 2 * lhi;
      v2f a;
      a.x = sA[16 * wr + lrow][ka];
      a.y = sA[16 * wr + lrow][ka + 1];
#pragma unroll
      for (int cc = 0; cc < 4; ++cc) {
        v2f bf;
        bf.x = sBT[64 * wc + 16 * cc + lrow][ka];
        bf.y = sBT[64 * wc + 16 * cc + lrow][ka + 1];
        acc[cc] = WMMA_F32(a, bf, acc[cc]);
      }
    }
    __syncthreads();
  }
  const int grow_base = row0 + 16 * wr + 8 * lhi;
  const int gcol      = col0 + 64 * wc + lrow;
#pragma unroll
  for (int r = 0; r < 8; ++r) {
    const int grow = grow_base + r;
#pragma unroll
    for (int cc = 0; cc < 4; ++cc) {
      const int c = gcol + 16 * cc;
      const size_t o = (size_t)grow * SEQ + c;
      float v = acc[cc][r] * SCALE + P[o];
      if (c > grow) v = NEGMAX;
      S[o] = v;
    }
  }
}

// ---------------------------------------------------------------------------
// Kernel 3: per-row softmax statistics (max, 1/sum(exp)). One block per row.
// ---------------------------------------------------------------------------
__global__ __launch_bounds__(256) void softmax_stats(
    const float* __restrict__ sim, float* __restrict__ stats) {
  const int row = blockIdx.x;                  // 0 .. B*SEQ-1
  const float* S = sim + (size_t)row * SEQ;
  __shared__ float red[256];
  const int t = threadIdx.x;

  float m = NEGMAX;
  for (int j = t; j < SEQ; j += 256) m = fmaxf(m, S[j]);
  red[t] = m;
  __syncthreads();
  for (int s = 128; s > 0; s >>= 1) {
    if (t < s) red[t] = fmaxf(red[t], red[t + s]);
    __syncthreads();
  }
  m = red[0];
  __syncthreads();

  float sum = 0.f;
  for (int j = t; j < SEQ; j += 256) sum += expf(S[j] - m);
  red[t] = sum;
  __syncthreads();
  for (int s = 128; s > 0; s >>= 1) {
    if (t < s) red[t] += red[t + s];
    __syncthreads();
  }
  if (t == 0) {
    stats[(size_t)row * 2 + 0] = m;
    stats[(size_t)row * 2 + 1] = 1.0f / red[0];
  }
}

// ---------------------------------------------------------------------------
// Kernel 4: tmp = softmax(sim) @ V. Attn probs computed on the fly while
// staging the A tile (exp(s-max)*invsum). K-loop stops at the causal edge.
// Per batch: M=4096, N=512, K<=row0+64.
// ---------------------------------------------------------------------------
__global__ __launch_bounds__(256) void pv_gemm(
    const float* __restrict__ sim, const float* __restrict__ stats,
    const float* __restrict__ Vb, float* __restrict__ Tmp) {
  const int b    = blockIdx.z;
  const int row0 = blockIdx.y * 64;    // i
  const int col0 = blockIdx.x * 128;   // d
  const float* S  = sim   + (size_t)b * SEQ * SEQ;
  const float* V  = Vb    + (size_t)b * SEQ * DIM;
  const float* st = stats + (size_t)b * SEQ * 2;
  float*       T  = Tmp   + (size_t)b * SEQ * DIM;

  __shared__ __align__(16) float sA[64][LP];
  __shared__ __align__(16) float sBT[128][LP];
  const int t    = threadIdx.x;
  const int lane = t & 31;
  const int wave = t >> 5;
  const int wr   = wave >> 1;
  const int wc   = wave & 1;
  const int lrow = lane & 15;
  const int lhi  = lane >> 4;

  const int ar = t >> 2;          // A tile 64x16 (prob rows)
  const int ac = (t & 3) * 4;
  const int kB = t >> 4;          // B tile: 16 j x 128 d -> transposed
  const int nB = (t & 15) * 8;

  const float mx = st[(size_t)(row0 + ar) * 2 + 0];
  const float is = st[(size_t)(row0 + ar) * 2 + 1];

  v8f acc[4] = {};
  const int kmax = row0 + 64;     // j > row0+63 contributes zero (causal)
  for (int k0 = 0; k0 < kmax; k0 += 16) {
    float4 s4 = *(const float4*)(S + (size_t)(row0 + ar) * SEQ + k0 + ac);
    sA[ar][ac + 0] = expf(s4.x - mx) * is;
    sA[ar][ac + 1] = expf(s4.y - mx) * is;
    sA[ar][ac + 2] = expf(s4.z - mx) * is;
    sA[ar][ac + 3] = expf(s4.w - mx) * is;
    float4 v0 = *(const float4*)(V + (size_t)(k0 + kB) * DIM + col0 + nB);
    float4 v1 = *(const float4*)(V + (size_t)(k0 + kB) * DIM + col0 + nB + 4);
    sBT[nB + 0][kB] = v0.x; sBT[nB + 1][kB] = v0.y;
    sBT[nB + 2][kB] = v0.z; sBT[nB + 3][kB] = v0.w;
    sBT[nB + 4][kB] = v1.x; sBT[nB + 5][kB] = v1.y;
    sBT[nB + 6][kB] = v1.z; sBT[nB + 7][kB] = v1.w;
    __syncthreads();
#pragma unroll
    for (int kk = 0; kk < 16; kk += 4) {
      const int ka = kk + 2 * lhi;
      v2f a;
      a.x = sA[16 * wr + lrow][ka];
      a.y = sA[16 * wr + lrow][ka + 1];
#pragma unroll
      for (int cc = 0; cc < 4; ++cc) {
        v2f bf;
        bf.x = sBT[64 * wc + 16 * cc + lrow][ka];
        bf.y = sBT[64 * wc + 16 * cc + lrow][ka + 1];
        acc[cc] = WMMA_F32(a, bf, acc[cc]);
      }
    }
    __syncthreads();
  }
  const int grow_base = row0 + 16 * wr + 8 * lhi;
  const int gcol      = col0 + 64 * wc + lrow;
#pragma unroll
  for (int r = 0; r < 8; ++r) {
    const size_t off = (size_t)(grow_base + r) * DIM;
#pragma unroll
    for (int cc = 0; cc < 4; ++cc) T[off + gcol + 16 * cc] = acc[cc][r];
  }
}

// ---------------------------------------------------------------------------
// Kernel 5: out = tmp @ Wout + bout.  M=16384, N=K=512.
// ---------------------------------------------------------------------------
__global__ __launch_bounds__(256) void out_gemm(
    const float* __restrict__ A, const float* __restrict__ W,
    const float* __restrict__ bias, float* __restrict__ Out) {
  __shared__ __align__(16) float sA[64][LP];
  __shared__ __align__(16) float sBT[128][LP];
  const int row0 = blockIdx.y * 64;
  const int col0 = blockIdx.x * 128;
  const int t    = threadIdx.x;
  const int lane = t & 31;
  const int wave = t >> 5;
  const int wr   = wave >> 1;
  const int wc   = wave & 1;
  const int lrow = lane & 15;
  const int lhi  = lane >> 4;

  const int ar = t >> 2;
  const int ac = (t & 3) * 4;
  const int kB = t >> 4;
  const int nB = (t & 15) * 8;

  v8f acc[4] = {};
  for (int k0 = 0; k0 < DIM; k0 += 16) {
    float4 va = *(const float4*)(A + (size_t)(row0 + ar) * DIM + k0 + ac);
    *(float4*)&sA[ar][ac] = va;
    float4 w0 = *(const float4*)(W + (size_t)(k0 + kB) * DIM + col0 + nB);
    float4 w1 = *(const float4*)(W + (size_t)(k0 + kB) * DIM + col0 + nB + 4);
    sBT[nB + 0][kB] = w0.x; sBT[nB + 1][kB] = w0.y;
    sBT[nB + 2][kB] = w0.z; sBT[nB + 3][kB] = w0.w;
    sBT[nB + 4][kB] = w1.x; sBT[nB + 5][kB] = w1.y;
    sBT[nB + 6][kB] = w1.z; sBT[nB + 7][kB] = w1.w;
    __syncthreads();
#pragma unroll
    for (int kk = 0; kk < 16; kk += 4) {
      const int ka = kk + 2 * lhi;
      v2f a;
      a.x = sA[16 * wr + lrow][ka];
      a.y = sA[16 * wr + lrow][ka + 1];
#pragma unroll
      for (int cc = 0; cc < 4; ++cc) {
        v2f bf;
        bf.x = sBT[64 * wc + 16 * cc + lrow][ka];
        bf.y = sBT[64 * wc + 16 * cc + lrow][ka + 1];
        acc[cc] = WMMA_F32(a, bf, acc[cc]);
      }
    }
    __syncthreads();
  }
  const int grow_base = row0 + 16 * wr + 8 * lhi;
  const int gcol      = col0 + 64 * wc + lrow;
  float bv[4];
#pragma unroll
  for (int cc = 0; cc < 4; ++cc) bv[cc] = bias[gcol + 16 * cc];
#pragma unroll
  for (int r = 0; r < 8; ++r) {
    const size_t off = (size_t)(grow_base + r) * DIM;
#pragma unroll
    for (int cc = 0; cc < 4; ++cc) Out[off + gcol + 16 * cc] = acc[cc][r] + bv[cc];
  }
}

// ---------------------------------------------------------------------------
// Launch
// ---------------------------------------------------------------------------
extern "C" void kernel_launch(void* const* d_in, const int* in_sizes, int n_in,
                              void* d_out, int out_size, void* d_ws, size_t ws_size,
                              hipStream_t stream) {
  (void)in_sizes; (void)n_in; (void)out_size; (void)ws_size;
  const float* x    = (const float*)d_in[0];   // [4,4096,512]
  const float* prev = (const float*)d_in[1];   // [4,4096,4096]
  const float* Wqkv = (const float*)d_in[2];   // [512,1536]
  const float* Wout = (const float*)d_in[3];   // [512,512]
  const float* bout = (const float*)d_in[4];   // [512]

  float* out = (float*)d_out;                           // [4,4096,512]
  float* sim = out + (size_t)BATCH * SEQ * DIM;         // [4,4096,4096]

  const size_t mat = (size_t)BATCH * SEQ * DIM;
  float* Qb    = (float*)d_ws;
  float* Kb    = Qb + mat;
  float* Vb    = Kb + mat;
  float* stats = Vb + mat;                              // [B*SEQ][2]
  float* Tmp   = Qb;  // Q dead after sim_gemm; reuse its 32 MB for tmp.

  dim3 blk(256);
  qkv_gemm<<<dim3(THREE_D / 128, (BATCH * SEQ) / 64), blk, 0, stream>>>(
      x, Wqkv, Qb, Kb, Vb);
  sim_gemm<<<dim3(SEQ / 128, SEQ / 64, BATCH), blk, 0, stream>>>(
      Qb, Kb, prev, sim);
  softmax_stats<<<dim3(BATCH * SEQ), blk, 0, stream>>>(sim, stats);
  pv_gemm<<<dim3(DIM / 128, SEQ / 64, BATCH), blk, 0, stream>>>(
      sim, stats, Vb, Tmp);
  out_gemm<<<dim3(DIM / 128, (BATCH * SEQ) / 64), blk, 0, stream>>>(
      Tmp, Wout, bout, out);
}